// GALNLLLoss2d_48850958025371
// MI455X (gfx1250) — compile-verified
//
#include <hip/hip_runtime.h>
#include <hip/hip_bf16.h>
#include <math.h>

// CDNA5 / gfx1250: wave32, WMMA 16x16 shapes.
typedef __attribute__((ext_vector_type(2))) float f32x2;
typedef __attribute__((ext_vector_type(4))) float f32x4;
typedef __attribute__((ext_vector_type(8))) float f32x8;

#define NBLOCKS 1024
#define NTHREADS 256

// ---------------------------------------------------------------------------
// log K_1(x), Abramowitz & Stegun 9.8.7 / 9.8.8 polynomial fits (matches ref).
// Branch-free: compute both branches, select.
// ---------------------------------------------------------------------------
__device__ __forceinline__ float log_k1f(float x) {
    // ---- small branch (x <= 2) ----
    float xs = fminf(x, 2.0f);
    float t  = (xs * (1.0f / 3.75f)); t = t * t;
    float i1 = xs * (0.5f + t * (0.87890594f + t * (0.51498869f + t * (0.15084934f
             + t * (0.02658733f + t * (0.00301532f + t * 0.00032411f))))));
    float u  = (xs * 0.5f); u = u * u;
    float ps = 1.0f + u * (0.15443144f + u * (-0.67278579f + u * (-0.18156897f
             + u * (-0.01919402f + u * (-0.00110404f + u * (-0.00004686f))))));
    float k1s = __logf(xs * 0.5f) * i1 + ps / xs;
    float log_small = __logf(k1s);
    // ---- large branch (x > 2), computed in log space ----
    float xl = fmaxf(x, 2.0f);
    float v  = 2.0f / xl;
    float pl = 1.25331414f + v * (0.23498619f + v * (-0.03655620f + v * (0.01504268f
             + v * (-0.00780353f + v * (0.00325614f - v * 0.00068245f)))));
    float log_large = -xl - 0.5f * __logf(xl) + __logf(pl);
    return (x <= 2.0f) ? log_small : log_large;
}

// ---------------------------------------------------------------------------
// Block reduction of 256 thread-partials using V_WMMA_F32_16X16X4_F32.
// The 256 LDS floats are a 16x16 f32 matrix; with A = ones(16x4), four chained
// WMMAs (D = 1*B_i + C) yield column sums of the whole block (every D row is
// identical), so acc[0] holds colsum[lane%16] in every lane of wave 0.
// Four wave32 xor-shuffles finish 16 -> 1. Exact f32 RNE tree reduction.
// ---------------------------------------------------------------------------
__device__ __forceinline__ float block_reduce_wmma(float val, float* lds) {
    const int tid = threadIdx.x;
    lds[tid] = val;
    __syncthreads();
    float total = 0.0f;
    if (tid < 32) {                      // entire wave 0 -> EXEC all ones
        f32x2 ones; ones[0] = 1.0f; ones[1] = 1.0f;
        f32x8 acc = {};
#pragma unroll
        for (int i = 0; i < 4; ++i) {
            f32x2 b;
            b[0] = lds[i * 64 + tid];        // each of the 256 values lands in
            b[1] = lds[i * 64 + 32 + tid];   // exactly one B slot exactly once
            acc = __builtin_amdgcn_wmma_f32_16x16x4_f32(
                false, ones, false, b, (short)0, acc, false, false);
        }
        float s = acc[0];                // colsum[lane % 16], all lanes
        s += __shfl_xor(s, 1, 32);
        s += __shfl_xor(s, 2, 32);
        s += __shfl_xor(s, 4, 32);
        s += __shfl_xor(s, 8, 32);       // grand total in every lane
        total = s;
    }
    return total;
}

// ---------------------------------------------------------------------------
// Kernel 1: streaming per-row log-prob + block partial sums.
// 32 B/row, 64 MB total -> pure HBM-bandwidth bound (~2.7 us at 23.3 TB/s).
// Non-temporal b64/b128 loads; f32 TRANS ops co-execute with VALU.
// ---------------------------------------------------------------------------
__global__ void __launch_bounds__(NTHREADS)
gal_nll_partial(const float* __restrict__ m, const float* __restrict__ L,
                const float* __restrict__ x, float* __restrict__ partial, int n) {
    __shared__ float lds[NTHREADS];
    const int gid    = blockIdx.x * blockDim.x + threadIdx.x;
    const int stride = gridDim.x * blockDim.x;

    const float LN2    = 0.6931471805599453f;   // log(2)
    const float LN_2PI = 1.8378770664093453f;   // log(2*pi)

    float acc = 0.0f;
    for (int i = gid; i < n; i += stride) {
        f32x2 mm = __builtin_nontemporal_load(((const f32x2*)m) + i);
        f32x4 Lv = __builtin_nontemporal_load(((const f32x4*)L) + i);
        f32x2 xx = __builtin_nontemporal_load(((const f32x2*)x) + i);

        const float l00 = Lv[0], l10 = Lv[2], l11 = Lv[3];
        // S = L L^T (2x2 SPD), explicit inverse via adjugate / det
        const float S00 = l00 * l00;
        const float S01 = l00 * l10;
        const float S11 = l10 * l10 + l11 * l11;
        const float det = S00 * S11 - S01 * S01;        // = (l00*l11)^2 > 0
        const float inv_det = 1.0f / det;

        const float m0 = mm[0], m1 = mm[1], x0 = xx[0], x1 = xx[1];
        const float qm  = (S11 * m0 * m0 - 2.0f * S01 * m0 * m1 + S00 * m1 * m1) * inv_det;
        const float qx  = (S11 * x0 * x0 - 2.0f * S01 * x0 * x1 + S00 * x1 * x1) * inv_det;
        const float qxm = (S11 * x0 * m0 - S01 * (x0 * m1 + x1 * m0) + S00 * x1 * m1) * inv_det;

        const float a = 2.0f + qm;
        const float z = __fsqrt_rn(a * qx);
        // log 2 + x'S⁻¹m - log(2π) + ½ log det + ½ log(qx/a) + log K₁(z)
        const float lp = LN2 + qxm - LN_2PI
                       + 0.5f * __logf(det)
                       + 0.5f * (__logf(qx) - __logf(a))
                       + log_k1f(z);
        acc += lp;
    }

    const float bsum = block_reduce_wmma(acc, lds);
    if (threadIdx.x == 0) partial[blockIdx.x] = bsum;
}

// ---------------------------------------------------------------------------
// Kernel 2: reduce NBLOCKS partials with the same WMMA tree, emit -mean.
// Single block -> deterministic result every replay (no float atomics).
// ---------------------------------------------------------------------------
__global__ void __launch_bounds__(NTHREADS)
gal_nll_final(const float* __restrict__ partial, float* __restrict__ out,
              int nparts, float inv_n) {
    __shared__ float lds[NTHREADS];
    float acc = 0.0f;
    for (int i = threadIdx.x; i < nparts; i += NTHREADS) acc += partial[i];
    const float total = block_reduce_wmma(acc, lds);
    if (threadIdx.x == 0) out[0] = -total * inv_n;
}

// ---------------------------------------------------------------------------
extern "C" void kernel_launch(void* const* d_in, const int* in_sizes, int n_in,
                              void* d_out, int out_size, void* d_ws, size_t ws_size,
                              hipStream_t stream) {
    const float* m = (const float*)d_in[0];   // [B,2]
    const float* L = (const float*)d_in[1];   // [B,2,2]
    const float* x = (const float*)d_in[2];   // [B,2]
    const int n = in_sizes[0] / 2;            // B
    float* partial = (float*)d_ws;            // NBLOCKS floats (4 KB)

    gal_nll_partial<<<NBLOCKS, NTHREADS, 0, stream>>>(m, L, x, partial, n);
    gal_nll_final<<<1, NTHREADS, 0, stream>>>(partial, (float*)d_out,
                                              NBLOCKS, 1.0f / (float)n);
}